// LINEMF_fac_56556129354268
// MI455X (gfx1250) — compile-verified
//
#include <hip/hip_runtime.h>
#include <stdint.h>

typedef float v4f __attribute__((ext_vector_type(4)));

#define NDIM 8192
#define BNEG 5
#define CLAMP_MIN 0.001f
#define ROWS_PER_BLOCK 8
#define TPB 256

// ---------------------------------------------------------------------------
// Kernel 1: vol_G = sum(d_rt); write b[j] = 1/d[j], a[i] = (vol_G/B)/d[i]
// Single block, 1024 threads (32 wave32s), shuffle + LDS tree reduction.
// ---------------------------------------------------------------------------
__global__ __launch_bounds__(1024) void linemf_prep_kernel(
    const float* __restrict__ d_rt, float* __restrict__ ws_a,
    float* __restrict__ ws_b) {
  const int tid = threadIdx.x;  // 0..1023
  float inv[8];
  float s = 0.0f;
#pragma unroll
  for (int k = 0; k < 8; ++k) {
    const int idx = tid + k * 1024;
    const float d = d_rt[idx];
    const float iv = 1.0f / d;  // IEEE divide (matches jnp 1.0/d_rt)
    inv[k] = iv;
    ws_b[idx] = iv;
    s += d;
  }
  // wave32 butterfly reduce
#pragma unroll
  for (int off = 16; off > 0; off >>= 1) s += __shfl_xor(s, off, 32);

  __shared__ float red[32];
  __shared__ float scale_sh;
  const int wave = tid >> 5;
  const int lane = tid & 31;
  if (lane == 0) red[wave] = s;
  __syncthreads();
  if (wave == 0) {
    float t = red[lane];
#pragma unroll
    for (int off = 16; off > 0; off >>= 1) t += __shfl_xor(t, off, 32);
    if (lane == 0) scale_sh = t / (float)BNEG;  // vol_G / B
  }
  __syncthreads();
  const float scale = scale_sh;
#pragma unroll
  for (int k = 0; k < 8; ++k) {
    const int idx = tid + k * 1024;
    ws_a[idx] = scale * inv[k];
  }
}

// ---------------------------------------------------------------------------
// Kernel 2: out[i,j] = log(max(X[i,j] * a[i] * b[j], 1e-3))
// Each block: async-stage b[0..8191] (32 KB) into LDS once, reuse over 8 rows.
// X / out streamed with non-temporal 128-bit accesses (512 MB total; HBM-bound).
// ---------------------------------------------------------------------------
__global__ __launch_bounds__(TPB) void linemf_main_kernel(
    const float* __restrict__ X, const float* __restrict__ ws_a,
    const float* __restrict__ ws_b, float* __restrict__ out) {
  __shared__ __align__(16) float bsh[NDIM];  // 32 KB
  const int tid = threadIdx.x;  // 0..255

  // Stage b -> LDS with CDNA5 async copies (ASYNCcnt-tracked, bypasses VGPRs).
  // 8 passes x 256 lanes x 16B = 32 KB. Generic-pointer low 32 bits are the
  // wave-relative LDS byte offset (ISA 10.2 aperture truncation).
#pragma unroll
  for (int it = 0; it < 8; ++it) {
    const int f4 = tid + it * TPB;  // float4 index 0..2047
    const uint64_t gaddr = (uint64_t)(uintptr_t)(ws_b + 4 * f4);
    const uint32_t laddr = (uint32_t)(uintptr_t)(&bsh[4 * f4]);
    asm volatile("global_load_async_to_lds_b128 %0, %1, off"
                 :
                 : "v"(laddr), "v"(gaddr)
                 : "memory");
  }
  asm volatile("s_wait_asynccnt 0" ::: "memory");
  __syncthreads();  // make staged LDS visible across all 8 waves

  const int row0 = blockIdx.x * ROWS_PER_BLOCK;
#pragma unroll 1
  for (int r = 0; r < ROWS_PER_BLOCK; ++r) {
    const int row = row0 + r;
    const float a = ws_a[row];  // block-uniform, L0-resident
    const v4f* __restrict__ xrow = (const v4f*)(X + (size_t)row * NDIM);
    v4f* __restrict__ orow = (v4f*)(out + (size_t)row * NDIM);
#pragma unroll
    for (int it = 0; it < 8; ++it) {
      const int f4 = tid + it * TPB;
      const v4f x = __builtin_nontemporal_load(xrow + f4);  // th:NT stream
      const v4f b = *(const v4f*)(&bsh[4 * f4]);            // ds_load_b128
      v4f y;
      y.x = __builtin_logf(fmaxf(x.x * a * b.x, CLAMP_MIN));
      y.y = __builtin_logf(fmaxf(x.y * a * b.y, CLAMP_MIN));
      y.z = __builtin_logf(fmaxf(x.z * a * b.z, CLAMP_MIN));
      y.w = __builtin_logf(fmaxf(x.w * a * b.w, CLAMP_MIN));
      __builtin_nontemporal_store(y, orow + f4);  // th:NT write-once
    }
  }
}

extern "C" void kernel_launch(void* const* d_in, const int* in_sizes, int n_in,
                              void* d_out, int out_size, void* d_ws,
                              size_t ws_size, hipStream_t stream) {
  (void)in_sizes;
  (void)n_in;
  (void)out_size;
  (void)ws_size;
  const float* X = (const float*)d_in[0];     // (8192, 8192) fp32
  const float* d_rt = (const float*)d_in[1];  // (8192,) fp32
  float* out = (float*)d_out;                 // (8192, 8192) fp32

  float* ws_a = (float*)d_ws;   // 8192 floats: (vol_G/B) * d^-1
  float* ws_b = ws_a + NDIM;    // 8192 floats: d^-1

  linemf_prep_kernel<<<1, 1024, 0, stream>>>(d_rt, ws_a, ws_b);
  linemf_main_kernel<<<NDIM / ROWS_PER_BLOCK, TPB, 0, stream>>>(X, ws_a, ws_b,
                                                                out);
}